// Net_37589553774905
// MI455X (gfx1250) — compile-verified
//
#include <hip/hip_runtime.h>

typedef __attribute__((ext_vector_type(2))) float v2f;
typedef __attribute__((ext_vector_type(8))) float v8f;

#define EPS 1e-5f

__device__ __forceinline__ float2 cmul(float2 a, float2 b) {
    return make_float2(a.x * b.x - a.y * b.y, a.x * b.y + a.y * b.x);
}
__device__ __forceinline__ float2 cfma(float2 a, float2 b, float2 c) {
    c.x += a.x * b.x - a.y * b.y;
    c.y += a.x * b.y + a.y * b.x;
    return c;
}
// log_magnitude: log1p(|x|) * exp(i*angle(x)) == x * log1p(r)/r
__device__ __forceinline__ float2 logmag(float2 v) {
    float r = sqrtf(v.x * v.x + v.y * v.y);
    float s = (r > 1e-30f) ? (log1pf(r) / r) : 1.0f;
    return make_float2(v.x * s, v.y * s);
}

// ---------------------------------------------------------------------------
// Stage 0 (fused): conv1 (Ci=3, per-pixel weights) + inst_norm + log_mag +
// exact 2x2 avg-pool 50->25, one block per (b, oc). Entire 50x50 channel image
// (20KB) lives in LDS — avoids materializing the 41MB conv1 output in HBM.
// ---------------------------------------------------------------------------
__global__ __launch_bounds__(256) void k_conv1_norm_pool(
    const float2* __restrict__ x,   // (32,3,2500)
    const float2* __restrict__ w,   // (64,3,2500)
    const float2* __restrict__ g, const float2* __restrict__ be,  // (64)
    float2* __restrict__ out)       // (32,64,625)
{
    __shared__ float2 tile[2500];
    __shared__ float2 redc[256];
    __shared__ float  redf[256];
    const int b = blockIdx.x >> 6;
    const int o = blockIdx.x & 63;
    const int tid = threadIdx.x;
    const float2* xb = x + (size_t)b * 3 * 2500;
    const float2* wo = w + (size_t)o * 3 * 2500;

    float2 lsum = make_float2(0.f, 0.f);
    for (int p = tid; p < 2500; p += 256) {
        float2 acc = make_float2(0.f, 0.f);
        acc = cfma(xb[p],          wo[p],          acc);
        acc = cfma(xb[2500 + p],   wo[2500 + p],   acc);
        acc = cfma(xb[5000 + p],   wo[5000 + p],   acc);
        tile[p] = acc;
        lsum.x += acc.x; lsum.y += acc.y;
    }
    redc[tid] = lsum;
    __syncthreads();
    for (int s = 128; s > 0; s >>= 1) {
        if (tid < s) { redc[tid].x += redc[tid + s].x; redc[tid].y += redc[tid + s].y; }
        __syncthreads();
    }
    const float2 m = make_float2(redc[0].x * (1.f / 2500.f), redc[0].y * (1.f / 2500.f));
    float lvar = 0.f;
    for (int p = tid; p < 2500; p += 256) {
        float dx = tile[p].x - m.x, dy = tile[p].y - m.y;
        lvar += dx * dx + dy * dy;
    }
    redf[tid] = lvar;
    __syncthreads();
    for (int s = 128; s > 0; s >>= 1) {
        if (tid < s) redf[tid] += redf[tid + s];
        __syncthreads();
    }
    const float sc = rsqrtf(redf[0] * (1.f / 2500.f) + EPS);
    const float2 ga = g[o], bb = be[o];
    for (int p = tid; p < 2500; p += 256) {
        float2 xn = make_float2((tile[p].x - m.x) * sc, (tile[p].y - m.y) * sc);
        float2 y = cmul(ga, xn);
        y.x += bb.x; y.y += bb.y;
        tile[p] = logmag(y);
    }
    __syncthreads();
    // exact 2x2 average pool 50 -> 25
    for (int p = tid; p < 625; p += 256) {
        int oh = p / 25, ow = p % 25;
        const float2* t = &tile[oh * 100 + ow * 2];
        float2 s = make_float2((t[0].x + t[1].x + t[50].x + t[51].x) * 0.25f,
                               (t[0].y + t[1].y + t[50].y + t[51].y) * 0.25f);
        out[(size_t)blockIdx.x * 625 + p] = s;
    }
}

// ---------------------------------------------------------------------------
// Generic instance-norm (+ optional per-channel spatial pre-multiply, optional
// post log-magnitude). One block per (b,c); channel image staged in LDS.
// Safe in-place (block owns its slice).
// ---------------------------------------------------------------------------
__global__ __launch_bounds__(256) void k_norm(
    const float2* __restrict__ in, float2* __restrict__ out,
    const float2* __restrict__ premul,  // (C,HW) or null
    const float2* __restrict__ g, const float2* __restrict__ be,  // (C)
    int C, int HW, int do_logmag)
{
    __shared__ float2 tile[640];   // max HW = 625
    __shared__ float2 redc[256];
    __shared__ float  redf[256];
    const int c = blockIdx.x % C;
    const size_t base = (size_t)blockIdx.x * HW;
    const int tid = threadIdx.x;

    float2 lsum = make_float2(0.f, 0.f);
    for (int p = tid; p < HW; p += 256) {
        float2 v = in[base + p];
        if (premul) v = cmul(v, premul[(size_t)c * HW + p]);
        tile[p] = v;
        lsum.x += v.x; lsum.y += v.y;
    }
    redc[tid] = lsum;
    __syncthreads();
    for (int s = 128; s > 0; s >>= 1) {
        if (tid < s) { redc[tid].x += redc[tid + s].x; redc[tid].y += redc[tid + s].y; }
        __syncthreads();
    }
    const float inv = 1.f / (float)HW;
    const float2 m = make_float2(redc[0].x * inv, redc[0].y * inv);
    float lvar = 0.f;
    for (int p = tid; p < HW; p += 256) {
        float dx = tile[p].x - m.x, dy = tile[p].y - m.y;
        lvar += dx * dx + dy * dy;
    }
    redf[tid] = lvar;
    __syncthreads();
    for (int s = 128; s > 0; s >>= 1) {
        if (tid < s) redf[tid] += redf[tid + s];
        __syncthreads();
    }
    const float sc = rsqrtf(redf[0] * inv + EPS);
    const float2 ga = g[c], bb = be[c];
    for (int p = tid; p < HW; p += 256) {
        float2 xn = make_float2((tile[p].x - m.x) * sc, (tile[p].y - m.y) * sc);
        float2 y = cmul(ga, xn);
        y.x += bb.x; y.y += bb.y;
        if (do_logmag) y = logmag(y);
        out[base + p] = y;
    }
}

// ---------------------------------------------------------------------------
// Per-pixel complex GEMM via V_WMMA_F32_16X16X4_F32.
// Out[b, n, pix] = sum_k X[b, k, pix] * W[n, k, wpix]   (complex)
// One wave per (pixel, 16-wide N tile, 16-wide M(batch) tile).
// Complex product via 4 real-WMMA accumulators: Cr = RR - II, Ci = RI + IR.
// WHW==HW -> per-pixel weights (c1); WHW==1 -> 1x1 shortcut conv (sc_w).
// EXEC is all-ones (blockDim=32, no divergence) as WMMA requires.
// ---------------------------------------------------------------------------
__global__ __launch_bounds__(32) void k_cgemm_wmma(
    const float2* __restrict__ X,   // (32, Ci, HW)
    const float2* __restrict__ W,   // (Co, Ci, WHW)
    float2* __restrict__ Out,       // (32, Co, HW)
    int Ci, int Co, int HW, int WHW)
{
    const int pix   = blockIdx.x;
    const int nBase = blockIdx.y << 4;
    const int mBase = blockIdx.z << 4;
    const int lane  = threadIdx.x;
    const int half  = lane >> 4;     // 0: K pair (k0,k0+1), 1: (k0+2,k0+3)
    const int l15   = lane & 15;
    const int wpix  = (WHW == 1) ? 0 : pix;

    v8f accRR = {}; v8f accII = {}; v8f accRI = {}; v8f accIR = {};

    for (int k0 = 0; k0 < Ci; k0 += 4) {
        const int ka = k0 + (half << 1);
        // A (16x4, M=batch rows): lane holds A[m][ka], A[m][ka+1]
        const float2* ap = X + ((size_t)(mBase + l15) * Ci + ka) * HW + pix;
        const float2 a0 = ap[0];
        const float2 a1 = ap[HW];
        v2f ar = { a0.x, a1.x };
        v2f ai = { a0.y, a1.y };
        // B (4x16, N=out-channel cols): lane holds B[ka][n], B[ka+1][n]
        const float2* bp = W + ((size_t)(nBase + l15) * Ci + ka) * WHW + wpix;
        const float2 b0 = bp[0];
        const float2 b1 = bp[WHW];
        v2f br = { b0.x, b1.x };
        v2f bi = { b0.y, b1.y };

        accRR = __builtin_amdgcn_wmma_f32_16x16x4_f32(false, ar, false, br, (short)0, accRR, false, false);
        accII = __builtin_amdgcn_wmma_f32_16x16x4_f32(false, ai, false, bi, (short)0, accII, false, false);
        accRI = __builtin_amdgcn_wmma_f32_16x16x4_f32(false, ar, false, bi, (short)0, accRI, false, false);
        accIR = __builtin_amdgcn_wmma_f32_16x16x4_f32(false, ai, false, br, (short)0, accIR, false, false);
    }

    // D 16x16: VGPR v -> row v (lanes 0-15) / row v+8 (lanes 16-31), col = l15
    #pragma unroll
    for (int v = 0; v < 8; ++v) {
        const int m = mBase + v + (half << 3);
        const int n = nBase + l15;
        float2 r;
        r.x = accRR[v] - accII[v];
        r.y = accRI[v] + accIR[v];
        Out[((size_t)m * Co + n) * HW + pix] = r;
    }
}

// Generic adaptive average pool (PyTorch start/end rule), thread per output.
__global__ void k_pool(const float2* __restrict__ in, float2* __restrict__ out,
                       int C, int IS, int OS)
{
    const int idx = blockIdx.x * blockDim.x + threadIdx.x;
    const int total = 32 * C * OS * OS;
    if (idx >= total) return;
    const int p  = idx % (OS * OS);
    const int bc = idx / (OS * OS);
    const int oh = p / OS, ow = p % OS;
    const int hs = (oh * IS) / OS, he = ((oh + 1) * IS + OS - 1) / OS;
    const int vs = (ow * IS) / OS, ve = ((ow + 1) * IS + OS - 1) / OS;
    float2 s = make_float2(0.f, 0.f);
    for (int h = hs; h < he; ++h)
        for (int w2 = vs; w2 < ve; ++w2) {
            float2 v = in[(size_t)bc * IS * IS + h * IS + w2];
            s.x += v.x; s.y += v.y;
        }
    const float inv = 1.f / (float)((he - hs) * (ve - vs));
    out[idx] = make_float2(s.x * inv, s.y * inv);
}

__global__ void k_add_logmag(const float2* __restrict__ a, const float2* __restrict__ b,
                             float2* __restrict__ out, int n)
{
    const int i = blockIdx.x * blockDim.x + threadIdx.x;
    if (i >= n) return;
    out[i] = logmag(make_float2(a[i].x + b[i].x, a[i].y + b[i].y));
}

// ComplexAdaptiveMaxPool2d(1): independent max over real and imag parts.
__global__ void k_maxpool(const float2* __restrict__ in, float2* __restrict__ out,
                          int C, int HW)
{
    const int i = blockIdx.x * blockDim.x + threadIdx.x;
    if (i >= 32 * C) return;
    float mr = -3.4e38f, mi = -3.4e38f;
    for (int p = 0; p < HW; ++p) {
        float2 v = in[(size_t)i * HW + p];
        mr = fmaxf(mr, v.x);
        mi = fmaxf(mi, v.y);
    }
    out[i] = make_float2(mr, mi);
}

__global__ void k_fc(const float2* __restrict__ h, const float2* __restrict__ w,
                     float2* __restrict__ out)
{
    const int i = blockIdx.x * blockDim.x + threadIdx.x;
    if (i >= 32 * 6) return;
    const int b = i / 6, o = i % 6;
    float2 acc = make_float2(0.f, 0.f);
    for (int c = 0; c < 512; ++c)
        acc = cfma(h[(size_t)b * 512 + c], w[(size_t)o * 512 + c], acc);
    out[i] = acc;
}

extern "C" void kernel_launch(void* const* d_in, const int* in_sizes, int n_in,
                              void* d_out, int out_size, void* d_ws, size_t ws_size,
                              hipStream_t stream) {
    (void)in_sizes; (void)n_in; (void)out_size; (void)ws_size;
    // complex64 buffers are interleaved {re,im} float pairs -> float2
    const float2* x      = (const float2*)d_in[0];
    const float2* conv1  = (const float2*)d_in[1];
    const float2* bn1_g  = (const float2*)d_in[2];
    const float2* bn1_b  = (const float2*)d_in[3];
    // l1 (block_same, 64ch, 25x25)
    const float2* l1_c1 = (const float2*)d_in[4];
    const float2* l1_g1 = (const float2*)d_in[5];
    const float2* l1_b1 = (const float2*)d_in[6];
    const float2* l1_c2 = (const float2*)d_in[7];
    const float2* l1_g2 = (const float2*)d_in[8];
    const float2* l1_b2 = (const float2*)d_in[9];
    // l2 (block_diff 64->128, 12x12)
    const float2* l2_c1 = (const float2*)d_in[10];
    const float2* l2_g1 = (const float2*)d_in[11];
    const float2* l2_b1 = (const float2*)d_in[12];
    const float2* l2_c2 = (const float2*)d_in[13];
    const float2* l2_g2 = (const float2*)d_in[14];
    const float2* l2_b2 = (const float2*)d_in[15];
    const float2* l2_sw = (const float2*)d_in[16];
    const float2* l2_sg = (const float2*)d_in[17];
    const float2* l2_sb = (const float2*)d_in[18];
    // l3 (128->256, 6x6)
    const float2* l3_c1 = (const float2*)d_in[19];
    const float2* l3_g1 = (const float2*)d_in[20];
    const float2* l3_b1 = (const float2*)d_in[21];
    const float2* l3_c2 = (const float2*)d_in[22];
    const float2* l3_g2 = (const float2*)d_in[23];
    const float2* l3_b2 = (const float2*)d_in[24];
    const float2* l3_sw = (const float2*)d_in[25];
    const float2* l3_sg = (const float2*)d_in[26];
    const float2* l3_sb = (const float2*)d_in[27];
    // l4 (256->512, 3x3)
    const float2* l4_c1 = (const float2*)d_in[28];
    const float2* l4_g1 = (const float2*)d_in[29];
    const float2* l4_b1 = (const float2*)d_in[30];
    const float2* l4_c2 = (const float2*)d_in[31];
    const float2* l4_g2 = (const float2*)d_in[32];
    const float2* l4_b2 = (const float2*)d_in[33];
    const float2* l4_sw = (const float2*)d_in[34];
    const float2* l4_sg = (const float2*)d_in[35];
    const float2* l4_sb = (const float2*)d_in[36];
    const float2* fcw   = (const float2*)d_in[37];

    char* ws = (char*)d_ws;
    float2* buf1 = (float2*)(ws);                        // 12 MB cap
    float2* buf2 = (float2*)(ws + (size_t)(12 << 20));   // 12 MB cap
    float2* buf3 = (float2*)(ws + (size_t)(24 << 20));   //  6 MB cap

    // Stage 0: conv1 + IN + logmag + pool 50->25  -> buf1 (32,64,25,25)
    k_conv1_norm_pool<<<dim3(32 * 64), 256, 0, stream>>>(x, conv1, bn1_g, bn1_b, buf1);

    // l1 block_same (identity shortcut)
    k_norm<<<32 * 64, 256, 0, stream>>>(buf1, buf2, l1_c1, l1_g1, l1_b1, 64, 625, 1);
    k_norm<<<32 * 64, 256, 0, stream>>>(buf2, buf2, l1_c2, l1_g2, l1_b2, 64, 625, 0);
    int n = 32 * 64 * 625;
    k_add_logmag<<<(n + 255) / 256, 256, 0, stream>>>(buf2, buf1, buf2, n);

    // pool 25->12 -> buf1 (32,64,12,12)
    n = 32 * 64 * 144;
    k_pool<<<(n + 255) / 256, 256, 0, stream>>>(buf2, buf1, 64, 25, 12);

    // l2 block_diff 64->128, HW=144
    k_cgemm_wmma<<<dim3(144, 128 / 16, 2), 32, 0, stream>>>(buf1, l2_c1, buf2, 64, 128, 144, 144);
    k_norm<<<32 * 128, 256, 0, stream>>>(buf2, buf2, nullptr, l2_g1, l2_b1, 128, 144, 1);
    k_norm<<<32 * 128, 256, 0, stream>>>(buf2, buf2, l2_c2, l2_g2, l2_b2, 128, 144, 0);
    k_cgemm_wmma<<<dim3(144, 128 / 16, 2), 32, 0, stream>>>(buf1, l2_sw, buf3, 64, 128, 144, 1);
    k_norm<<<32 * 128, 256, 0, stream>>>(buf3, buf3, nullptr, l2_sg, l2_sb, 128, 144, 0);
    n = 32 * 128 * 144;
    k_add_logmag<<<(n + 255) / 256, 256, 0, stream>>>(buf2, buf3, buf2, n);

    // pool 12->6 -> buf1 (32,128,6,6)
    n = 32 * 128 * 36;
    k_pool<<<(n + 255) / 256, 256, 0, stream>>>(buf2, buf1, 128, 12, 6);

    // l3 block_diff 128->256, HW=36
    k_cgemm_wmma<<<dim3(36, 256 / 16, 2), 32, 0, stream>>>(buf1, l3_c1, buf2, 128, 256, 36, 36);
    k_norm<<<32 * 256, 256, 0, stream>>>(buf2, buf2, nullptr, l3_g1, l3_b1, 256, 36, 1);
    k_norm<<<32 * 256, 256, 0, stream>>>(buf2, buf2, l3_c2, l3_g2, l3_b2, 256, 36, 0);
    k_cgemm_wmma<<<dim3(36, 256 / 16, 2), 32, 0, stream>>>(buf1, l3_sw, buf3, 128, 256, 36, 1);
    k_norm<<<32 * 256, 256, 0, stream>>>(buf3, buf3, nullptr, l3_sg, l3_sb, 256, 36, 0);
    n = 32 * 256 * 36;
    k_add_logmag<<<(n + 255) / 256, 256, 0, stream>>>(buf2, buf3, buf2, n);

    // pool 6->3 -> buf1 (32,256,3,3)
    n = 32 * 256 * 9;
    k_pool<<<(n + 255) / 256, 256, 0, stream>>>(buf2, buf1, 256, 6, 3);

    // l4 block_diff 256->512, HW=9
    k_cgemm_wmma<<<dim3(9, 512 / 16, 2), 32, 0, stream>>>(buf1, l4_c1, buf2, 256, 512, 9, 9);
    k_norm<<<32 * 512, 256, 0, stream>>>(buf2, buf2, nullptr, l4_g1, l4_b1, 512, 9, 1);
    k_norm<<<32 * 512, 256, 0, stream>>>(buf2, buf2, l4_c2, l4_g2, l4_b2, 512, 9, 0);
    k_cgemm_wmma<<<dim3(9, 512 / 16, 2), 32, 0, stream>>>(buf1, l4_sw, buf3, 256, 512, 9, 1);
    k_norm<<<32 * 512, 256, 0, stream>>>(buf3, buf3, nullptr, l4_sg, l4_sb, 512, 9, 0);
    n = 32 * 512 * 9;
    k_add_logmag<<<(n + 255) / 256, 256, 0, stream>>>(buf2, buf3, buf2, n);

    // complex max-pool (3x3 -> 1) then FC (32x512)x(512x6)
    k_maxpool<<<(32 * 512 + 255) / 256, 256, 0, stream>>>(buf2, buf3, 512, 9);
    k_fc<<<1, 192, 0, stream>>>(buf3, fcw, (float2*)d_out);
}